// TcSpace_91302414778869
// MI455X (gfx1250) — compile-verified
//
#include <hip/hip_runtime.h>
#include <math.h>

typedef __attribute__((ext_vector_type(2))) float v2f;
typedef __attribute__((ext_vector_type(8))) float v8f;

#define T_ 4
#define B_ 512
#define G_ 16
#define N_ 256
#define D_ 32
#define FRAME_STRIDE (B_ * N_ * D_)   // 4194304 floats between frames of z_what
#define CELLSTRIDE   (N_)             // per (t,b) presence stride

// -------- deterministic block reduction (256 threads) --------
__device__ __forceinline__ float block_reduce(float v, float* red) {
    int tid = threadIdx.x;
    red[tid] = v;
    __syncthreads();
    for (int s = 128; s > 0; s >>= 1) {
        if (tid < s) red[tid] += red[tid + s];
        __syncthreads();
    }
    float r = red[0];
    __syncthreads();
    return r;
}

// -------- kernel 1: z_what_loss = sum((zw[1:]-zw[:-1])^2) --------
__global__ void zwhat_kernel(const float* __restrict__ zw, float* __restrict__ P1) {
    __shared__ float red[256];
    const float4* z4 = (const float4*)zw;
    size_t gid = (size_t)blockIdx.x * 256 + threadIdx.x;
    float acc = 0.f;
    // 3 * 512*256*32 / 4 = 3,145,728 float4 pairs; per-t stride = 2^20 float4
    for (size_t f = gid; f < 3145728u; f += 1024u * 256u) {
        size_t t = f >> 20, rem = f & 1048575u;
        float4 x = z4[((t + 1) << 20) + rem];
        float4 y = z4[(t << 20) + rem];
        float dx = x.x - y.x, dy = x.y - y.y, dz = x.z - y.z, dw = x.w - y.w;
        acc += dx * dx + dy * dy + dz * dz + dw * dw;
    }
    float r = block_reduce(acc, red);
    if (threadIdx.x == 0) P1[blockIdx.x] = r;
}

// -------- kernel 2: z_pres_loss + objects_detected --------
__global__ void pres_kernel(const float* __restrict__ zpres,
                            const float* __restrict__ zpp,
                            float* __restrict__ P2a, float* __restrict__ P2b) {
    __shared__ float red[256];
    const int S = B_ * N_;  // 131072
    int idx = blockIdx.x * 256 + threadIdx.x;
    float z0 = zpres[idx], z1 = zpres[S + idx], z2 = zpres[2 * S + idx], z3 = zpres[3 * S + idx];
    float acc = 0.f;
    {   float sim = 1.f - (z2 - z0) * (z2 - z0);
        float de  = (z2 - z1) * (z2 - z1) + (z0 - z1) * (z0 - z1);
        acc += sim * de; }
    {   float sim = 1.f - (z3 - z1) * (z3 - z1);
        float de  = (z3 - z2) * (z3 - z2) + (z1 - z2) * (z1 - z2);
        acc += sim * de; }
    float cnt = (zpp[idx] > 0.5f ? 1.f : 0.f)
              + (zpp[S + idx] > 0.5f ? 1.f : 0.f)
              + (zpp[2 * S + idx] > 0.5f ? 1.f : 0.f);
    float a = block_reduce(acc, red);
    float b = block_reduce(cnt, red);
    if (threadIdx.x == 0) { P2a[blockIdx.x] = a; P2b[blockIdx.x] = b; }
}

// -------- kernel 3: pool_loss --------
// one block per (t,b), t in 0..2. 256 threads = one per cell.
__global__ void pool_kernel(const float* __restrict__ zw,
                            const float* __restrict__ zpp,
                            float* __restrict__ P3) {
    __shared__ float sa[16][256];   // [d-half][cell]  zw_w of frame t
    __shared__ float red[256];
    int tid = threadIdx.x, bx = blockIdx.x;
    int t = bx / B_, b = bx - t * B_;
    const float* Pt = zw + (size_t)(t * B_ + b) * (N_ * D_);
    const float* Pn = Pt + FRAME_STRIDE;
    float ppt = zpp[(t * B_ + b) * N_ + tid];
    float pp1 = zpp[((t + 1) * B_ + b) * N_ + tid];
    int i = tid >> 4, j = tid & 15;
    float dot = 0.f, na = 0.f, nb2 = 0.f;
    for (int h = 0; h < 2; h++) {
        const float4* p4 = (const float4*)(Pt + tid * D_ + h * 16);
        #pragma unroll
        for (int e = 0; e < 4; e++) {
            float4 v = p4[e];
            sa[e * 4 + 0][tid] = fabsf(v.x) * ppt;
            sa[e * 4 + 1][tid] = fabsf(v.y) * ppt;
            sa[e * 4 + 2][tid] = fabsf(v.z) * ppt;
            sa[e * 4 + 3][tid] = fabsf(v.w) * ppt;
        }
        float bw[16];
        const float4* q4 = (const float4*)(Pn + tid * D_ + h * 16);
        #pragma unroll
        for (int e = 0; e < 4; e++) {
            float4 v = q4[e];
            bw[e * 4 + 0] = fabsf(v.x) * pp1;
            bw[e * 4 + 1] = fabsf(v.y) * pp1;
            bw[e * 4 + 2] = fabsf(v.z) * pp1;
            bw[e * 4 + 3] = fabsf(v.w) * pp1;
        }
        __syncthreads();
        #pragma unroll
        for (int dd = 0; dd < 16; dd++) {
            float mx = sa[dd][tid];  // center always in-window; zw_w >= 0
            for (int di = -1; di <= 1; di++) {
                int ii = i + di;
                if (ii < 0 || ii > 15) continue;
                for (int dj = -1; dj <= 1; dj++) {
                    int jj = j + dj;
                    if (jj < 0 || jj > 15) continue;
                    mx = fmaxf(mx, sa[dd][ii * 16 + jj]);
                }
            }
            float bb = bw[dd];
            dot += mx * bb; na += mx * mx; nb2 += bb * bb;
        }
        __syncthreads();
    }
    float cosim = dot / fmaxf(sqrtf(na) * sqrtf(nb2), 1e-6f);
    float contrib = -(cosim * 0.5f * (ppt + pp1));
    float r = block_reduce(contrib, red);
    if (threadIdx.x == 0) P3[bx] = r;
}

// -------- kernel 4: objects_loss via f32 WMMA Gram bands --------
// one block per (t,b). 8 waves; each wave handles prior rows i = wave, wave+8.
// For each (prior row i, next row i2 = i+di mod 16): Gram16x16 = A(16x32) * B(32x16)
// via 8 chained V_WMMA_F32_16X16X4_F32, then extract the +/-1 band (toroidal).
__global__ void objects_kernel(const float* __restrict__ zw,
                               const float* __restrict__ zpp,
                               float* __restrict__ P4) {
    __shared__ float sp[256];        // prior norms
    __shared__ float sn[256];        // next norms
    __shared__ float pres[256];      // next presence prob
    __shared__ float det[256];       // prior detected (0/1)
    __shared__ float tile[8][16 * 17]; // per-wave Gram tile (padded)
    __shared__ float red[256];
    int tid = threadIdx.x, bx = blockIdx.x;
    int t = bx / B_, b = bx - t * B_;
    const float* Pt = zw + (size_t)(t * B_ + b) * (N_ * D_);
    const float* Pn = Pt + FRAME_STRIDE;
    {
        const float4* p4 = (const float4*)(Pt + tid * D_);
        float s = 0.f;
        #pragma unroll
        for (int e = 0; e < 8; e++) { float4 v = p4[e]; s += v.x*v.x + v.y*v.y + v.z*v.z + v.w*v.w; }
        sp[tid] = sqrtf(s);
        const float4* q4 = (const float4*)(Pn + tid * D_);
        s = 0.f;
        #pragma unroll
        for (int e = 0; e < 8; e++) { float4 v = q4[e]; s += v.x*v.x + v.y*v.y + v.z*v.z + v.w*v.w; }
        sn[tid] = sqrtf(s);
        pres[tid] = zpp[((t + 1) * B_ + b) * N_ + tid];
        det[tid]  = (zpp[(t * B_ + b) * N_ + tid] > 0.5f) ? 1.f : 0.f;
    }
    __syncthreads();

    int wave = tid >> 5, lane = tid & 31;
    int n15 = lane & 15, hi = lane >> 4;
    float lowm = (lane < 16) ? 1.f : 0.f;  // upper half lanes are layout duplicates
    float obj = 0.f;

    for (int rep = 0; rep < 2; rep++) {
        int i = wave + rep * 8;
        // A tile (prior row i), documented f32 16x4 layout, float2 per lane per K-chunk
        v2f a[8];
        const float* Arow = Pt + (size_t)(i * 16 + n15) * D_ + 2 * hi;
        #pragma unroll
        for (int kc = 0; kc < 8; kc++) a[kc] = *(const v2f*)(Arow + kc * 4);

        float ssum = 0.f, smax = -1e30f, anym = 0.f;
        float pn_j = sp[i * 16 + n15];

        for (int di = -1; di <= 1; di++) {
            int i2 = (i + di) & 15;
            const float* Brow = Pn + (size_t)(i2 * 16 + n15) * D_ + 2 * hi;
            v8f c = {};
            #pragma unroll
            for (int kc = 0; kc < 8; kc++) {
                v2f bv = *(const v2f*)(Brow + kc * 4);
                c = __builtin_amdgcn_wmma_f32_16x16x4_f32(
                        false, a[kc], false, bv, (short)0, c, false, false);
            }
            // spill Gram tile: VGPR r, lane L -> (M = r + 8*(L>=16), N = L&15)
            int row0 = hi * 8;
            #pragma unroll
            for (int r = 0; r < 8; r++)
                tile[wave][(row0 + r) * 17 + n15] = c[r];
            __asm__ volatile("s_wait_dscnt 0" ::: "memory");
            // band extraction for prior cell (i, j=n15)
            #pragma unroll
            for (int dj = -1; dj <= 1; dj++) {
                int jn = (n15 + dj) & 15;
                float g = tile[wave][n15 * 17 + jn];
                int cell2 = i2 * 16 + jn;
                float s = g / fmaxf(pn_j * sn[cell2], 1e-8f);
                float m = (pres[cell2] > 0.5f) ? 1.f : 0.f;
                ssum += m * s;
                smax = fmaxf(smax, (m > 0.f) ? s : -1e30f);
                anym = fmaxf(anym, m);
            }
        }
        float per = ssum - 5.0f * smax;
        float valid = lowm * anym * det[i * 16 + n15];
        obj += valid * per;
    }
    float r = block_reduce(obj, red);
    if (threadIdx.x == 0) P4[bx] = r;
}

// -------- kernel 5: deterministic final reduction + loss composition --------
__global__ void finalize_kernel(const float* __restrict__ ws,
                                const float* __restrict__ base,
                                const int* __restrict__ gstep,
                                float* __restrict__ out) {
    __shared__ float red[256];
    int tid = threadIdx.x;
    auto sumArr = [&](const float* p, int n) -> float {
        float v = 0.f;
        for (int k = tid; k < n; k += 256) v += p[k];
        return block_reduce(v, red);
    };
    float zwl  = sumArr(ws + 0,    1024);
    float zpl  = sumArr(ws + 1024, 512);
    float nobj = sumArr(ws + 1536, 512);
    float pl   = sumArr(ws + 2048, 1536);
    float ol   = sumArr(ws + 3584, 1536);
    if (tid == 0) {
        float bsum = base[0] + base[1] + base[2] + base[3];
        float scaling = fminf(1.0f, (float)gstep[0] / 300000.0f);
        float loss = bsum + zwl * 10.0f + zpl * 1.0f + pl * 1.0f + ol * scaling * 10.0f;
        out[0] = loss; out[1] = zwl; out[2] = zpl;
        out[3] = pl;   out[4] = ol;  out[5] = nobj;
    }
}

extern "C" void kernel_launch(void* const* d_in, const int* in_sizes, int n_in,
                              void* d_out, int out_size, void* d_ws, size_t ws_size,
                              hipStream_t stream) {
    const float* zw   = (const float*)d_in[0];  // z_what (4,512,256,32)
    const float* zpp  = (const float*)d_in[1];  // z_pres_prob (4,512,256,1)
    const float* zp   = (const float*)d_in[2];  // z_pres (4,512,256,1)
    const float* base = (const float*)d_in[3];  // base_losses (4,)
    const int*  gstep = (const int*)d_in[4];    // global_step scalar
    float* ws  = (float*)d_ws;                  // partials: [0,1024) zwl | [1024,1536) zpl
                                                // [1536,2048) nobj | [2048,3584) pool | [3584,5120) obj
    float* out = (float*)d_out;

    zwhat_kernel  <<<1024, 256, 0, stream>>>(zw, ws + 0);
    pres_kernel   <<<512,  256, 0, stream>>>(zp, zpp, ws + 1024, ws + 1536);
    pool_kernel   <<<1536, 256, 0, stream>>>(zw, zpp, ws + 2048);
    objects_kernel<<<1536, 256, 0, stream>>>(zw, zpp, ws + 3584);
    finalize_kernel<<<1,   256, 0, stream>>>(ws, base, gstep, out);
}